// SimMetricBlock_39204461478656
// MI455X (gfx1250) — compile-verified
//
#include <hip/hip_runtime.h>

// MI455X / gfx1250, wave32. Store-bound cosine-similarity GEMM:
//   phase 1: Xn = normalize(x*w) rows (to d_ws, 4 MB)
//   phase 2: out[b] = threshold(Xn[b] . Xn[b]^T) via V_WMMA_F32_16X16X4_F32

typedef __attribute__((ext_vector_type(2))) float v2f;
typedef __attribute__((ext_vector_type(8))) float v8f;

#define NDIM 4096
#define FDIM 64
#define BDIM 4
#define EMP_T 0.1f
#define ZERO_V 9e-15f

// ---------------- Phase 1: row normalization ----------------
// One wave per row (64 floats -> float2 per lane). 8 waves per block.
__global__ __launch_bounds__(256) void norm_rows(const float* __restrict__ x,
                                                 const float* __restrict__ w,
                                                 float* __restrict__ xn) {
  const int row  = blockIdx.x * 8 + (threadIdx.x >> 5);   // [0, B*N)
  const int lane = threadIdx.x & 31;
  const size_t xoff = (size_t)row * FDIM + (size_t)(lane << 1);
  const size_t woff = (size_t)(row & (NDIM - 1)) * FDIM + (size_t)(lane << 1);

  v2f xv = *(const v2f*)(x + xoff);
  v2f wv = *(const v2f*)(w + woff);
  float p0 = xv.x * wv.x;
  float p1 = xv.y * wv.y;

  float ss = p0 * p0 + p1 * p1;
#pragma unroll
  for (int off = 1; off < 32; off <<= 1)
    ss += __shfl_xor(ss, off, 32);

  const float inv = 1.0f / fmaxf(sqrtf(ss), 1e-8f);
  v2f o;
  o.x = p0 * inv;
  o.y = p1 * inv;
  *(v2f*)(xn + xoff) = o;
}

// ---------------- Phase 2: C = Xn . Xn^T with threshold epilogue ----------------
// Fragment fetch per the CDNA5 f32 16x4 A layout:
//   lane L: row = L&15, K pair base = 2*(L>>4); VGPR0/1 = K, K+1 (contiguous float2).
// B = Xn^T uses the transpose-mirror striping, so the B fragment is the same
// fetch pattern applied to the n-block rows of Xn.
__device__ __forceinline__ v2f load_frag(const float* __restrict__ p, int kb) {
  return *(const v2f*)(p + kb);
}

__device__ __forceinline__ void store_tile(float* __restrict__ outb, const v8f acc,
                                           int row0, int col0, int lane) {
  const int c    = lane & 15;
  const int rofs = (lane >> 4) << 3;   // lanes 16-31 hold rows +8
  float* p = outb + (size_t)(row0 + rofs) * NDIM + (size_t)(col0 + c);
#pragma unroll
  for (int v = 0; v < 8; ++v) {
    float val = acc[v];
    val = (val > EMP_T) ? val : ZERO_V;
    __builtin_nontemporal_store(val, p + (size_t)v * NDIM);
  }
}

__global__ __launch_bounds__(256) void cos_gemm(const float* __restrict__ xn,
                                                float* __restrict__ out) {
  const int lane = threadIdx.x & 31;
  const int wave = threadIdx.x >> 5;
  const int b    = blockIdx.z;

  // 8 waves as 2(M) x 4(N); wave tile 32x32; workgroup tile 64x128.
  const int m0 = blockIdx.y * 64  + (wave >> 2) * 32;
  const int n0 = blockIdx.x * 128 + (wave & 3)  * 32;

  const float* __restrict__ Xb = xn + (size_t)b * NDIM * FDIM;

  const int r    = lane & 15;
  const int koff = (lane >> 4) << 1;

  const float* pa0 = Xb + (size_t)(m0 + r)      * FDIM + koff;
  const float* pa1 = Xb + (size_t)(m0 + 16 + r) * FDIM + koff;
  const float* pb0 = Xb + (size_t)(n0 + r)      * FDIM + koff;
  const float* pb1 = Xb + (size_t)(n0 + 16 + r) * FDIM + koff;

  v8f acc00 = {};
  v8f acc01 = {};
  v8f acc10 = {};
  v8f acc11 = {};

#pragma unroll
  for (int ks = 0; ks < 16; ++ks) {
    const int kb = ks << 2;  // K advances 4 per WMMA
    v2f a0 = load_frag(pa0, kb);
    v2f a1 = load_frag(pa1, kb);
    v2f b0 = load_frag(pb0, kb);
    v2f b1 = load_frag(pb1, kb);
    acc00 = __builtin_amdgcn_wmma_f32_16x16x4_f32(false, a0, false, b0, (short)0, acc00, false, false);
    acc01 = __builtin_amdgcn_wmma_f32_16x16x4_f32(false, a0, false, b1, (short)0, acc01, false, false);
    acc10 = __builtin_amdgcn_wmma_f32_16x16x4_f32(false, a1, false, b0, (short)0, acc10, false, false);
    acc11 = __builtin_amdgcn_wmma_f32_16x16x4_f32(false, a1, false, b1, (short)0, acc11, false, false);
  }

  float* outb = out + (size_t)b * NDIM * NDIM;
  store_tile(outb, acc00, m0,      n0,      lane);
  store_tile(outb, acc01, m0,      n0 + 16, lane);
  store_tile(outb, acc10, m0 + 16, n0,      lane);
  store_tile(outb, acc11, m0 + 16, n0 + 16, lane);
}

// ---------------- Launch ----------------
extern "C" void kernel_launch(void* const* d_in, const int* in_sizes, int n_in,
                              void* d_out, int out_size, void* d_ws, size_t ws_size,
                              hipStream_t stream) {
  const float* x = (const float*)d_in[0];   // (4, 4096, 64)
  const float* w = (const float*)d_in[1];   // (1, 4096, 64)
  float* out = (float*)d_out;               // (4, 4096, 4096)
  float* xn  = (float*)d_ws;                // 4*4096*64 f32 = 4 MB scratch

  // Phase 1: 16384 rows, 8 rows per 256-thread block.
  norm_rows<<<dim3((BDIM * NDIM) / 8), dim3(256), 0, stream>>>(x, w, xn);

  // Phase 2: grid (N/128, N/64, B).
  cos_gemm<<<dim3(NDIM / 128, NDIM / 64, BDIM), dim3(256), 0, stream>>>(xn, out);
}